// Depth_CA_87497073755006
// MI455X (gfx1250) — compile-verified
//
#include <hip/hip_runtime.h>
#include <cmath>
#include <algorithm>

// ---------------------------------------------------------------------------
// Depth-dependent coded-aperture pipeline for MI455X (gfx1250).
// 2D FFTs implemented as radix-16 four-step transforms on the WMMA pipes:
//   256-pt FFT  =  F16 * M  -> twiddle ->  (.) * F16      (complex, f32)
// using V_WMMA_F32_16X16X4_F32 (4 chained k-steps per 16x16x16 matmul).
// Global->LDS staging uses GLOBAL_LOAD_ASYNC_TO_LDS_B64 (ASYNCcnt) when the
// toolchain exposes the builtin; falls back to synchronous loads otherwise.
// fftshift/ifftshift folded into XOR-128 addressing. wave32: one FFT per wave.
// ---------------------------------------------------------------------------

typedef __attribute__((ext_vector_type(2))) float v2f;
typedef __attribute__((ext_vector_type(8))) float v8f;
typedef __attribute__((ext_vector_type(2))) int   v2i;

#define SL 65536            // 256*256 elements per slice
#define NSL_PSF 45          // 15 depths * 3 bands
#define NSL_IMG 12          // 4 batch * 3 bands

#if __has_builtin(__builtin_amdgcn_global_load_async_to_lds_b64) && \
    __has_builtin(__builtin_amdgcn_s_wait_asynccnt)
#define USE_ASYNC_LDS 1
typedef __attribute__((address_space(1))) v2i gv2i_t;   // global b64 payload
typedef __attribute__((address_space(3))) v2i lv2i_t;   // LDS b64 payload
#endif

struct Coefs {
    float coef[45];   // per (depth,band) PHASE1 coefficient
    float p2c[3];     // per band PHASE2 coefficient
    float absScale2;  // |amplitude scale|^2 for psf
};

__device__ __forceinline__ v8f wmma4(v2f a, v2f b, v8f c) {
    // D = A(16x4,f32) * B(4x16,f32) + C(16x16,f32)
    return __builtin_amdgcn_wmma_f32_16x16x4_f32(
        false, a, false, b, (short)0, c, false, false);
}

// ---------------------------------------------------------------------------
// One 256-point complex FFT per wave, radix-16 x radix-16 via WMMA.
// colMajor=0: transform rows (contiguous), colMajor=1: transform columns.
// xLoad*/xStore*: XOR-128 (fftshift) on transform axis (T) / line axis (L).
// dir=+1 forward (exp(-i..)), dir=-1 inverse; scale applied at store.
// ---------------------------------------------------------------------------
__global__ __launch_bounds__(256) void k_fft256(
    const float2* __restrict__ src, float2* __restrict__ dst,
    int nslices, int dir, int colMajor,
    int xLoadT, int xLoadL, int xStoreT, int xStoreL, float scale)
{
    __shared__ float2 tile[8][256];

    const int lane = threadIdx.x & 31;
    const int wave = threadIdx.x >> 5;
    const int lineG = blockIdx.x * 8 + wave;      // global line id
    const int s  = lineG >> 8;                    // slice
    const int j  = lineG & 255;                   // line within slice
    const float2* sp = src + (size_t)s * SL;
    float2*       dp = dst + (size_t)s * SL;

    const int xlt = xLoadT ? 128 : 0;
    const int jl  = j ^ (xLoadL ? 128 : 0);

    __builtin_prefetch(sp, 0, 0);                 // global_prefetch_b8

    // ---- stage 256 complex into per-wave LDS tile ----
#pragma unroll
    for (int i = 0; i < 8; ++i) {
        int t  = lane + 32 * i;
        int tt = t ^ xlt;
        int idx = colMajor ? (tt * 256 + jl) : (jl * 256 + tt);
#ifdef USE_ASYNC_LDS
        __builtin_amdgcn_global_load_async_to_lds_b64(
            (gv2i_t*)(sp + idx),
            (lv2i_t*)&tile[wave][t],
            0, 0);
#else
        tile[wave][t] = sp[idx];
#endif
    }
#ifdef USE_ASYNC_LDS
    __builtin_amdgcn_s_wait_asynccnt(0);
#endif
    __syncthreads();

    const float dsign = (dir > 0) ? -1.0f : 1.0f; // forward: exp(-i*...)
    const float PI2   = 6.283185307179586f;
    const int r  = lane & 15;                     // row (stage1) / col id
    const int hi = (lane >> 4) & 1;               // half-wave selector

    // ---- DFT16 fragments: F16[r][c] == F16[c][r]; reused by both stages ----
    v2f fAr[4], fAi[4], fAin[4];
#pragma unroll
    for (int kk = 0; kk < 4; ++kk) {
        int c0 = 4 * kk + 2 * hi;
        int c1 = c0 + 1;
        float a0s, a0c, a1s, a1c;
        __sincosf(dsign * PI2 * (float)((r * c0) & 15) * (1.0f / 16.0f), &a0s, &a0c);
        __sincosf(dsign * PI2 * (float)((r * c1) & 15) * (1.0f / 16.0f), &a1s, &a1c);
        fAr[kk]  = v2f{ a0c,  a1c };
        fAi[kk]  = v2f{ a0s,  a1s };
        fAin[kk] = v2f{ -a0s, -a1s };
    }

    // ---- stage 1:  Y = F16 * M   (M[n1][n0] = x[16*n1+n0]) ----
    v8f yr = {0,0,0,0,0,0,0,0};
    v8f yi = {0,0,0,0,0,0,0,0};
#pragma unroll
    for (int kk = 0; kk < 4; ++kk) {
        int c0 = 4 * kk + 2 * hi;                 // B rows (n1)
        int c1 = c0 + 1;
        float2 m0 = tile[wave][16 * c0 + r];
        float2 m1 = tile[wave][16 * c1 + r];
        v2f Br = { m0.x, m1.x };
        v2f Bi = { m0.y, m1.y };
        yr = wmma4(fAr[kk],  Br, yr);
        yr = wmma4(fAin[kk], Bi, yr);
        yi = wmma4(fAr[kk],  Bi, yi);
        yi = wmma4(fAi[kk],  Br, yi);
    }

    // ---- twiddle: Y'[k1][n0] *= W256^(k1*n0) ----
#pragma unroll
    for (int v = 0; v < 8; ++v) {
        int row = v + 8 * hi;                     // k1
        float ts, tc;
        __sincosf(dsign * PI2 * (float)((row * r) & 255) * (1.0f / 256.0f), &ts, &tc);
        float ar = yr[v], ai = yi[v];
        yr[v] = ar * tc - ai * ts;
        yi[v] = ar * ts + ai * tc;
    }
    __syncthreads();

    // ---- spill Y' back to LDS for the operand-role swap ----
#pragma unroll
    for (int v = 0; v < 8; ++v) {
        int row = v + 8 * hi;
        tile[wave][16 * row + r] = make_float2(yr[v], yi[v]);
    }
    __syncthreads();

    // ---- stage 2:  Z = Y' * F16  (B frag == stage-1 A frag, F16 symmetric) ----
    v8f zr = {0,0,0,0,0,0,0,0};
    v8f zi = {0,0,0,0,0,0,0,0};
#pragma unroll
    for (int kk = 0; kk < 4; ++kk) {
        int c0 = 4 * kk + 2 * hi;                 // A cols (n0)
        int c1 = c0 + 1;
        float2 y0 = tile[wave][16 * r + c0];
        float2 y1 = tile[wave][16 * r + c1];
        v2f Ar  = { y0.x, y1.x };
        v2f Ai  = { y0.y, y1.y };
        v2f Ain = { -Ai[0], -Ai[1] };
        zr = wmma4(Ar,  fAr[kk], zr);
        zr = wmma4(Ain, fAi[kk], zr);
        zi = wmma4(Ar,  fAi[kk], zi);
        zi = wmma4(Ai,  fAr[kk], zi);
    }

    // ---- store: out[k1 + 16*k0] = Z[k1][k0]; lane holds 8 consecutive k ----
    const int xst = xStoreT ? 128 : 0;
    const int jsl = j ^ (xStoreL ? 128 : 0);
#pragma unroll
    for (int v = 0; v < 8; ++v) {
        int k   = v + 8 * hi + 16 * r;
        int kk2 = k ^ xst;
        int idx = colMajor ? (kk2 * 256 + jsl) : (jsl * 256 + kk2);
        dp[idx] = make_float2(zr[v] * scale, zi[v] * scale);
    }
}

// ---------------------------------------------------------------------------
// Pointwise / reduction kernels
// ---------------------------------------------------------------------------
__global__ __launch_bounds__(256) void k_build_ph(float2* out, const float* H, Coefs cf) {
    int idx = blockIdx.x * 256 + threadIdx.x;     // 45*65536
    int sl = idx >> 16, p = idx & 65535;
    int y = p >> 8, x = p & 255;
    int d = sl / 3, band = sl - 3 * d;
    float ux = -0.004f + x * 3.125e-5f;           // u = -L_LEN/2 + i*DU
    float uy = -0.004f + y * 3.125e-5f;
    float xy = ux * ux + uy * uy;
    float rad = (xy <= 4.0e-6f) ? 1.0f : 0.0f;    // RADII^2
    float ca  = H[(y >> 4) * 16 + (x >> 4)];      // kron 16x16 tiling
    float ph  = cf.coef[d * 3 + band] * xy;
    float sn, cs; __sincosf(ph, &sn, &cs);
    float a = rad * ca;
    out[idx] = make_float2(a * cs, a * sn);
}

__global__ __launch_bounds__(256) void k_mul_phase2(float2* buf, Coefs cf) {
    int idx = blockIdx.x * 256 + threadIdx.x;     // 45*65536
    int sl = idx >> 16, p = idx & 65535;
    int y = p >> 8, x = p & 255;
    int band = sl % 3;
    float fx = (float)(((x + 128) & 255) - 128) * 125.0f;  // 1/L_LEN
    float fy = (float)(((y + 128) & 255) - 128) * 125.0f;
    float th = -cf.p2c[band] * (fx * fx + fy * fy);
    float sn, cs; __sincosf(th, &sn, &cs);
    float2 v = buf[idx];
    buf[idx] = make_float2(v.x * cs - v.y * sn, v.x * sn + v.y * cs);
}

__global__ __launch_bounds__(256) void k_psf_abs(const float2* vu, float* ps, float s2) {
    int idx = blockIdx.x * 256 + threadIdx.x;
    float2 v = vu[idx];
    ps[idx] = (v.x * v.x + v.y * v.y) * s2;       // unit phases drop under |.|^2
}

__global__ __launch_bounds__(256) void k_slice_sum(const float* ps, float* sums) {
    __shared__ float sm[256];
    const float* p = ps + (size_t)blockIdx.x * SL;
    float acc = 0.0f;
    for (int i = threadIdx.x; i < SL; i += 256) acc += p[i];
    sm[threadIdx.x] = acc; __syncthreads();
    for (int s = 128; s > 0; s >>= 1) {
        if (threadIdx.x < s) sm[threadIdx.x] += sm[threadIdx.x + s];
        __syncthreads();
    }
    if (threadIdx.x == 0) sums[blockIdx.x] = sm[0];
}

__global__ __launch_bounds__(256) void k_psf_div(float* ps, const float* sums) {
    int idx = blockIdx.x * 256 + threadIdx.x;
    ps[idx] /= sums[idx >> 16];
}

__global__ __launch_bounds__(256) void k_r2c(const float* in, float2* out, int n) {
    int idx = blockIdx.x * 256 + threadIdx.x;
    if (idx < n) out[idx] = make_float2(in[idx], 0.0f);
}

__global__ __launch_bounds__(256) void k_img2c(const float* img, float2* out) {
    int idx = blockIdx.x * 256 + threadIdx.x;     // 12*65536
    int sl = idx >> 16, p = idx & 65535;
    int b = sl / 3, band = sl % 3;
    out[idx] = make_float2(img[((size_t)b * SL + p) * 3 + band], 0.0f);
}

__global__ __launch_bounds__(256) void k_zero(float* p, int n) {
    int idx = blockIdx.x * 256 + threadIdx.x;
    if (idx < n) p[idx] = 0.0f;
}

__global__ __launch_bounds__(256) void k_blurmul(const float2* imf, const float2* pfr,
                                                 float2* outb, int d) {
    int idx = blockIdx.x * 256 + threadIdx.x;     // 12*65536
    int sl = idx >> 16, p = idx & 65535;
    int band = sl % 3;
    float2 a = imf[idx];
    float2 w = pfr[(size_t)(d * 3 + band) * SL + p];
    outb[idx] = make_float2(a.x * w.x - a.y * w.y, a.x * w.y + a.y * w.x);
}

__global__ __launch_bounds__(256) void k_accum(const float2* blur, const float* Map,
                                               float* res, int d) {
    int idx = blockIdx.x * 256 + threadIdx.x;     // 12*65536
    int sl = idx >> 16, p = idx & 65535;
    int b = sl / 3;
    float2 v = blur[idx];
    float mag = sqrtf(v.x * v.x + v.y * v.y);
    res[idx] += Map[((size_t)b * SL + p) * 15 + d] * mag;
}

__global__ __launch_bounds__(256) void k_set0(unsigned* p) { *p = 0u; }

__global__ __launch_bounds__(256) void k_maxred(const float* p, int n, unsigned* outm) {
    __shared__ float sm[256];
    float m = 0.0f;
    for (int i = blockIdx.x * 256 + threadIdx.x; i < n; i += gridDim.x * 256)
        m = fmaxf(m, p[i]);
    sm[threadIdx.x] = m; __syncthreads();
    for (int s = 128; s > 0; s >>= 1) {
        if (threadIdx.x < s) sm[threadIdx.x] = fmaxf(sm[threadIdx.x], sm[threadIdx.x + s]);
        __syncthreads();
    }
    if (threadIdx.x == 0) atomicMax(outm, __float_as_uint(sm[0])); // vals >= 0
}

__global__ __launch_bounds__(256) void k_divmax(float* p, int n, const unsigned* m) {
    int idx = blockIdx.x * 256 + threadIdx.x;
    if (idx < n) p[idx] /= __uint_as_float(*m);
}

__global__ __launch_bounds__(256) void k_wiener(const float2* pifr, const float2* rfr,
                                                const float2* pfr, float2* outb,
                                                int d, const float* param) {
    int idx = blockIdx.x * 256 + threadIdx.x;     // 12*65536
    int sl = idx >> 16, p = idx & 65535;
    int band = sl % 3;
    size_t ps = (size_t)(d * 3 + band) * SL + p;
    float2 a = pifr[ps];
    float2 r = rfr[idx];
    float2 w = pfr[ps];
    float inv = 1.0f / (w.x * w.x + w.y * w.y + param[0]);
    outb[idx] = make_float2((a.x * r.x - a.y * r.y) * inv,
                            (a.x * r.y + a.y * r.x) * inv);
}

__global__ __launch_bounds__(256) void k_recov(const float2* w, float* out, int d) {
    int idx = blockIdx.x * 256 + threadIdx.x;     // 12*65536
    int sl = idx >> 16, p = idx & 65535;
    int b = sl / 3, band = sl % 3;
    float2 v = w[idx];
    out[((size_t)b * SL + p) * 45 + band * 15 + d] = sqrtf(v.x * v.x + v.y * v.y);
}

// ---------------------------------------------------------------------------
// Host orchestration
// ---------------------------------------------------------------------------
static double deta_h(double l_um) {
    double l2 = l_um * l_um;
    return 0.5375 + 0.00829045 / l2 - 0.000211046 / (l2 * l2);
}

extern "C" void kernel_launch(void* const* d_in, const int* in_sizes, int n_in,
                              void* d_out, int out_size, void* d_ws, size_t ws_size,
                              hipStream_t stream) {
    const float* img   = (const float*)d_in[0];   // (4,256,256,3)
    const float* Map   = (const float*)d_in[1];   // (4,256,256,15)
    const float* H     = (const float*)d_in[2];   // (1,16,16,1)
    const float* param = (const float*)d_in[3];   // (1,)
    float* out = (float*)d_out;                   // (4,256,256,45)

    // ---- host constant math (double) ----
    const double ZI = 0.05, Z0 = 2.5;
    const double L_SEN = 6.22e-6 * 256.0, L_LEN = 0.008;
    const double DU = L_LEN / 256.0, DX2 = L_SEN / 256.0;
    const double F = 1.0 / (1.0 / ZI + 1.0 / Z0);
    const double Rf = F * deta_h(0.55);
    const double lamb[3] = { 460e-9, 550e-9, 640e-9 };
    const double flmb0 = Rf / deta_h(0.46);       // reference bug: band 0 everywhere
    double zs[15];
    for (int i = 0; i < 15; ++i)
        zs[i] = 8.0 - 3.0 * std::log(0.9 + i * (10.1 / 14.0));
    std::sort(zs, zs + 15);
    Coefs cf;
    for (int d = 0; d < 15; ++d)
        for (int b = 0; b < 3; ++b) {
            double K = 2.0 * M_PI / lamb[b];
            cf.coef[d * 3 + b] = (float)(-K / (2.0 * flmb0) + K / (2.0 * zs[d])
                + M_PI * (L_LEN - L_SEN) / (lamb[b] * ZI * L_LEN));
        }
    for (int b = 0; b < 3; ++b)
        cf.p2c[b] = (float)(M_PI * lamb[b] * ZI * L_LEN / L_SEN);
    double amp = (L_SEN / L_LEN) * (DU * DU) / (DX2 * DX2);
    cf.absScale2 = (float)(amp * amp);

    // ---- carve workspace (~120 MB; resident in 192 MB L2) ----
    char* base = (char*)d_ws;
    auto carve = [&](size_t bytes) {
        void* p = base; base += (bytes + 255) & ~(size_t)255; return p;
    };
    float2* A    = (float2*)carve((size_t)NSL_PSF * SL * 8);
    float2* Bb   = (float2*)carve((size_t)NSL_PSF * SL * 8);
    float2* PFR  = (float2*)carve((size_t)NSL_PSF * SL * 8);
    float2* PIFR = (float2*)carve((size_t)NSL_PSF * SL * 8);
    float*  PS   = (float*) carve((size_t)NSL_PSF * SL * 4);
    float2* IMF  = (float2*)carve((size_t)NSL_IMG * SL * 8);
    float2* RFR  = (float2*)carve((size_t)NSL_IMG * SL * 8);
    float*  RES  = (float*) carve((size_t)NSL_IMG * SL * 4);
    float*  SUMS = (float*) carve(64 * 4);
    unsigned* MAXR = (unsigned*)carve(64);
    unsigned* MAXO = (unsigned*)carve(64);

    auto fft = [&](const float2* s, float2* d, int ns, int dir, int cm,
                   int lt, int ll, int st, int sl2, float sc) {
        k_fft256<<<ns * 32, 256, 0, stream>>>(s, d, ns, dir, cm, lt, ll, st, sl2, sc);
    };
    const float is = 1.0f / 256.0f;
    const int GP = NSL_PSF * SL / 256;            // 11520 blocks
    const int GI = NSL_IMG * SL / 256;            // 3072 blocks

    // ---- PSF synthesis: ph -> fft2(fftshift(.)) -> *e^{-iP2} -> ifftshift(ifft2(.)) ----
    k_build_ph<<<GP, 256, 0, stream>>>(A, H, cf);
    fft(A, Bb, NSL_PSF, +1, 0, 1, 1, 0, 0, 1.0f);
    fft(Bb, A, NSL_PSF, +1, 1, 0, 0, 0, 0, 1.0f);
    k_mul_phase2<<<GP, 256, 0, stream>>>(A, cf);
    fft(A, Bb, NSL_PSF, -1, 0, 0, 0, 0, 0, is);
    fft(Bb, A, NSL_PSF, -1, 1, 0, 0, 1, 1, is);
    k_psf_abs<<<GP, 256, 0, stream>>>(A, PS, cf.absScale2);
    k_slice_sum<<<NSL_PSF, 256, 0, stream>>>(PS, SUMS);
    k_psf_div<<<GP, 256, 0, stream>>>(PS, SUMS);

    // ---- psf_fr = fft2c(psf);  psf_ifr = fftshift(ifft2(ifftshift(psf))) ----
    k_r2c<<<GP, 256, 0, stream>>>(PS, A, NSL_PSF * SL);
    fft(A, Bb, NSL_PSF, +1, 0, 1, 1, 0, 0, 1.0f);
    fft(Bb, PFR, NSL_PSF, +1, 1, 0, 0, 1, 1, 1.0f);
    k_r2c<<<GP, 256, 0, stream>>>(PS, A, NSL_PSF * SL);
    fft(A, Bb, NSL_PSF, -1, 0, 1, 1, 0, 0, is);
    fft(Bb, PIFR, NSL_PSF, -1, 1, 0, 0, 1, 1, is);

    // ---- img_ft = fft2c(img) ----
    k_img2c<<<GI, 256, 0, stream>>>(img, A);
    fft(A, Bb, NSL_IMG, +1, 0, 1, 1, 0, 0, 1.0f);
    fft(Bb, IMF, NSL_IMG, +1, 1, 0, 0, 1, 1, 1.0f);

    // ---- blur per depth, Map-weighted accumulation ----
    k_zero<<<GI, 256, 0, stream>>>(RES, NSL_IMG * SL);
    for (int d = 0; d < 15; ++d) {
        k_blurmul<<<GI, 256, 0, stream>>>(IMF, PFR, A, d);
        fft(A, Bb, NSL_IMG, -1, 0, 1, 1, 0, 0, is);
        fft(Bb, A, NSL_IMG, -1, 1, 0, 0, 1, 1, is);
        k_accum<<<GI, 256, 0, stream>>>(A, Map, RES, d);
    }
    k_set0<<<1, 1, 0, stream>>>(MAXR);
    k_maxred<<<256, 256, 0, stream>>>(RES, NSL_IMG * SL, MAXR);
    k_divmax<<<GI, 256, 0, stream>>>(RES, NSL_IMG * SL, MAXR);

    // ---- result_fr = fft2c(result) ----
    k_r2c<<<GI, 256, 0, stream>>>(RES, A, NSL_IMG * SL);
    fft(A, Bb, NSL_IMG, +1, 0, 1, 1, 0, 0, 1.0f);
    fft(Bb, RFR, NSL_IMG, +1, 1, 0, 0, 1, 1, 1.0f);

    // ---- Wiener deconvolution per depth ----
    for (int d = 0; d < 15; ++d) {
        k_wiener<<<GI, 256, 0, stream>>>(PIFR, RFR, PFR, A, d, param);
        fft(A, Bb, NSL_IMG, -1, 0, 1, 1, 0, 0, is);
        fft(Bb, A, NSL_IMG, -1, 1, 0, 0, 1, 1, is);
        k_recov<<<GI, 256, 0, stream>>>(A, out, d);
    }
    k_set0<<<1, 1, 0, stream>>>(MAXO);
    k_maxred<<<256, 256, 0, stream>>>(out, out_size, MAXO);
    k_divmax<<<(out_size + 255) / 256, 256, 0, stream>>>(out, out_size, MAXO);
}